// MPNN_47699906789974
// MI455X (gfx1250) — compile-verified
//
#include <hip/hip_runtime.h>
#include <hip/hip_bf16.h>

typedef float v2f __attribute__((ext_vector_type(2)));
typedef float v8f __attribute__((ext_vector_type(8)));

#define NN 100000
#define EE 1600000
#define GG 1000

// ---------------- utility: zero a float4 array ----------------
__global__ void zero_f4(float4* __restrict__ p, int n4) {
    int i = blockIdx.x * blockDim.x + threadIdx.x;
    if (i < n4) p[i] = make_float4(0.f, 0.f, 0.f, 0.f);
}

// ---------------- bincount(batch) -> counts[G] ----------------
__global__ void count_nodes(const int* __restrict__ batch, float* __restrict__ counts, int n) {
    int i = blockIdx.x * blockDim.x + threadIdx.x;
    if (i < n) atomicAdd(&counts[batch[i]], 1.0f);
}

// ---------------- edge gather/scale/scatter-add ----------------
// 16 threads per edge; each thread moves one float4 of the 64-float row.
// Gather is fully coalesced (consecutive lanes read consecutive float4 of a row);
// scatter uses no-return global_atomic_add_f32 that resolves in L2 (agg fits in 192MB L2).
__global__ void edge_scatter(const float* __restrict__ h,
                             const int* __restrict__ src,
                             const int* __restrict__ dst,
                             const float* __restrict__ ew,
                             float* __restrict__ agg, int ne) {
    int idx = blockIdx.x * blockDim.x + threadIdx.x;
    int e = idx >> 4;
    if (e >= ne) return;
    int c4 = (idx & 15) << 2;
    float w = ew[e];
    const float4 v = *(const float4*)(h + (size_t)src[e] * 64 + c4);
    float* a = agg + (size_t)dst[e] * 64 + c4;
    atomicAdd(a + 0, v.x * w);
    atomicAdd(a + 1, v.y * w);
    atomicAdd(a + 2, v.z * w);
    atomicAdd(a + 3, v.w * w);
}

// ---------------- out = act( (in * 1/cpn[row]) @ W[64x64] + b ) via V_WMMA_F32_16X16X4_F32 ----
// One wave per 16-row tile. K=64 -> 16 chained WMMAs per 16x16 output tile, 4 column tiles.
// Branches are wave-uniform so EXEC is all-ones at every WMMA (ISA requirement).
template <bool SIG>
__global__ void gemm64_wmma(const float* __restrict__ in,
                            const float* __restrict__ W,
                            const float* __restrict__ bias,
                            const int* __restrict__ batch,
                            const float* __restrict__ counts,
                            float* __restrict__ out, int n_tiles) {
    int wave = (blockIdx.x * blockDim.x + threadIdx.x) >> 5;
    int lane = threadIdx.x & 31;
    if (wave >= n_tiles) return;               // wave-uniform

    int m    = lane & 15;                      // row within tile (A: M = lane%16)
    int half = lane >> 4;                      // 0: K={0,1}, 1: K={2,3}
    int row0 = wave * 16;
    int row  = row0 + m;

    float inv = 1.0f / counts[batch[row]];

    // A fragments: per wmma step kk, lane holds in[row][kk*4 + 2*half + {0,1}] (contiguous float2)
    v2f a[16];
    const float* inrow = in + (size_t)row * 64 + 2 * half;
#pragma unroll
    for (int kk = 0; kk < 16; ++kk) {
        v2f t = *(const v2f*)(inrow + kk * 4);
        a[kk] = t * inv;                       // fold per-node normalization into A
    }

#pragma unroll
    for (int nt = 0; nt < 4; ++nt) {
        int n = nt * 16 + m;                   // B/C/D column for this lane
        // B fragments: B[k][n] with local K = v + 2*half
        v2f bb[16];
#pragma unroll
        for (int kk = 0; kk < 16; ++kk) {
            int k0 = kk * 4 + 2 * half;
            bb[kk].x = W[(size_t)(k0 + 0) * 64 + n];
            bb[kk].y = W[(size_t)(k0 + 1) * 64 + n];
        }
        float bv = bias[n];
        v8f c;
#pragma unroll
        for (int v = 0; v < 8; ++v) c[v] = bv; // C init = bias (each lane's 8 rows share column n)
#pragma unroll
        for (int kk = 0; kk < 16; ++kk)
            c = __builtin_amdgcn_wmma_f32_16x16x4_f32(
                    false, a[kk], false, bb[kk], (short)0, c, false, false);
        // D layout: VGPR v -> (M = v + 8*half, N = n)
#pragma unroll
        for (int v = 0; v < 8; ++v) {
            float val = c[v];
            if (SIG) val = 1.0f / (1.0f + __expf(-val));
            out[(size_t)(row0 + v + 8 * half) * 64 + n] = val;
        }
    }
}

// ---------------- layer 3 (64 -> 1) fused with global mean pool ----------------
// One wave per node: each lane dots a float2 slice, wave-reduce, lane0 atomically
// accumulates (dot*inv_cpn + b3) / counts[g] into pooled[g].
__global__ void layer3_pool(const float* __restrict__ agg,
                            const float* __restrict__ W3,
                            const float* __restrict__ b3,
                            const int* __restrict__ batch,
                            const float* __restrict__ counts,
                            float* __restrict__ pooled, int n) {
    int wave = (blockIdx.x * blockDim.x + threadIdx.x) >> 5;
    int lane = threadIdx.x & 31;
    if (wave >= n) return;
    int g = batch[wave];
    float inv = 1.0f / counts[g];
    float2 v = *(const float2*)(agg + (size_t)wave * 64 + 2 * lane);
    float2 w = *(const float2*)(W3 + 2 * lane);
    float s = (v.x * w.x + v.y * w.y) * inv;   // normalized dot contribution
#pragma unroll
    for (int o = 16; o > 0; o >>= 1) s += __shfl_xor(s, o, 32);
    if (lane == 0) atomicAdd(&pooled[g], (s + b3[0]) * inv);
}

extern "C" void kernel_launch(void* const* d_in, const int* in_sizes, int n_in,
                              void* d_out, int out_size, void* d_ws, size_t ws_size,
                              hipStream_t stream) {
    const float* x     = (const float*)d_in[0];
    const int*   ei    = (const int*)d_in[1];   // [2,E]: src = ei[0:E], dst = ei[E:2E]
    const float* ew    = (const float*)d_in[2];
    const int*   batch = (const int*)d_in[3];
    const float* W1 = (const float*)d_in[4];
    const float* b1 = (const float*)d_in[5];
    const float* W2 = (const float*)d_in[6];
    const float* b2 = (const float*)d_in[7];
    const float* W3 = (const float*)d_in[8];
    const float* b3 = (const float*)d_in[9];
    float* pooled = (float*)d_out;

    const int* src = ei;
    const int* dst = ei + EE;

    char*  ws     = (char*)d_ws;
    float* counts = (float*)ws;                                   // G floats
    float* bufA   = (float*)(ws + 4096);                          // N*64 floats
    float* bufB   = (float*)(ws + 4096 + (size_t)NN * 64 * 4);    // N*64 floats

    dim3 blk(256);
    const int nbig   = NN * 64 / 4;            // float4 count of a node buffer
    const int zb     = (nbig + 255) / 256;
    const int sb     = (EE * 16) / 256;        // edge-scatter blocks (exact)
    const int ntiles = NN / 16;                // 6250
    const int gb     = (ntiles + 7) / 8;       // 8 waves / block

    // counts + output init (out/ws are poisoned by the harness)
    zero_f4<<<1, blk, 0, stream>>>((float4*)counts, GG / 4);
    zero_f4<<<1, blk, 0, stream>>>((float4*)pooled, GG / 4);
    count_nodes<<<(NN + 255) / 256, blk, 0, stream>>>(batch, counts, NN);

    // layer 1
    zero_f4<<<zb, blk, 0, stream>>>((float4*)bufA, nbig);
    edge_scatter<<<sb, blk, 0, stream>>>(x, src, dst, ew, bufA, EE);
    gemm64_wmma<true><<<gb, blk, 0, stream>>>(bufA, W1, b1, batch, counts, bufB, ntiles);

    // layer 2
    zero_f4<<<zb, blk, 0, stream>>>((float4*)bufA, nbig);
    edge_scatter<<<sb, blk, 0, stream>>>(bufB, src, dst, ew, bufA, EE);
    gemm64_wmma<true><<<gb, blk, 0, stream>>>(bufA, W2, b2, batch, counts, bufB, ntiles);

    // layer 3 + pooling
    zero_f4<<<zb, blk, 0, stream>>>((float4*)bufA, nbig);
    edge_scatter<<<sb, blk, 0, stream>>>(bufB, src, dst, ew, bufA, EE);
    layer3_pool<<<(NN * 32 + 255) / 256, blk, 0, stream>>>(bufA, W3, b3, batch, counts, pooled, NN);
}